// DiffKS_73229192397157
// MI455X (gfx1250) — compile-verified
//
#include <hip/hip_runtime.h>

// Problem constants (from reference)
#define NB     48
#define T_LEN  88200
#define NE     6
#define NL     2

// Chunked-scan constants: 88200 = 252 chunks * 350 samples, 350 = 14 tiles * 25
#define L_CH   350
#define N_CH   252
#define TOT_CH (NB * N_CH)      // 12096 chunks total
#define CPB    128              // chunks (threads) per block
#define TT     25               // time steps per LDS tile
#define NTILE  (L_CH / TT)      // 14

#if __has_builtin(__builtin_amdgcn_global_load_async_to_lds_b32) && \
    __has_builtin(__builtin_amdgcn_global_load_async_to_lds_b64)
#define HAVE_ASYNC_LDS 1
#else
#define HAVE_ASYNC_LDS 0
#endif

// Types matching the builtin prototypes (per hipcc diagnostics):
//   b32: (as1 int*, as3 int*, imm int, imm int)
//   b64: (as1 v2i*, as3 v2i*, imm int, imm int)  with v2i = vector_size(8) int
typedef int v2i_t __attribute__((vector_size(8)));
typedef __attribute__((address_space(1))) int*   g_i32p;
typedef __attribute__((address_space(3))) int*   l_i32p;
typedef __attribute__((address_space(1))) v2i_t* g_v2ip;
typedef __attribute__((address_space(3))) v2i_t* l_v2ip;

__device__ __forceinline__ void wait_async_all() {
#if HAVE_ASYNC_LDS
#if __has_builtin(__builtin_amdgcn_s_wait_asynccnt)
  __builtin_amdgcn_s_wait_asynccnt(0);
#else
  asm volatile("s_wait_asynccnt 0" ::: "memory");
#endif
#endif
}

// ---------------------------------------------------------------------------
// K1: time-varying FIR (invert_lpc). Two samples per thread so the dominant
// A_exc stream (102 MB) is read as three exact float4 loads per thread.
//   x[t] = y[t] + sum_k A[t,k-1]*y[t-k]
// ---------------------------------------------------------------------------
__global__ __launch_bounds__(256) void k_fir(const float* __restrict__ y,
                                             const float* __restrict__ Aexc,
                                             float* __restrict__ x) {
  int p = blockIdx.x * blockDim.x + threadIdx.x;     // pair index
  if (p >= (NB * T_LEN) / 2) return;
  const int gi0 = 2 * p;                             // even flat index
  const int t0 = gi0 % T_LEN;                        // even sample index

  const float4* a4 = (const float4*)(Aexc + (size_t)gi0 * NE);
  float4 A0 = a4[0], A1 = a4[1], A2 = a4[2];

  float yw[8];                                       // y[t0-6 .. t0+1]
#pragma unroll
  for (int j = 0; j < 8; ++j) {
    int tt = t0 - 6 + j;
    yw[j] = (tt >= 0) ? y[gi0 - 6 + j] : 0.0f;
  }

  float acc0 = yw[6];
  acc0 = fmaf(A0.x, yw[5], acc0);
  acc0 = fmaf(A0.y, yw[4], acc0);
  acc0 = fmaf(A0.z, yw[3], acc0);
  acc0 = fmaf(A0.w, yw[2], acc0);
  acc0 = fmaf(A1.x, yw[1], acc0);
  acc0 = fmaf(A1.y, yw[0], acc0);

  float acc1 = yw[7];
  acc1 = fmaf(A1.z, yw[6], acc1);
  acc1 = fmaf(A1.w, yw[5], acc1);
  acc1 = fmaf(A2.x, yw[4], acc1);
  acc1 = fmaf(A2.y, yw[3], acc1);
  acc1 = fmaf(A2.z, yw[2], acc1);
  acc1 = fmaf(A2.w, yw[1], acc1);

  *(float2*)(x + gi0) = make_float2(acc0, acc1);
}

// ---------------------------------------------------------------------------
// K2 (REPLAY=false): per-chunk zero-state response + homogeneous responses.
//   summary[g] = { y0[L-1], y0[L-2], p1[L-1], p1[L-2], p2[L-1], p2[L-2] }
// K4 (REPLAY=true): per-chunk replay with true initial state, writes output.
// Double-buffered async global->LDS staging overlaps copy with the serial
// recurrence (the latency-bound part).
// ---------------------------------------------------------------------------
template <bool REPLAY>
__global__ __launch_bounds__(CPB) void k_chunks(const float* __restrict__ xbuf,
                                                const float* __restrict__ Aloop,
                                                float* __restrict__ summ,
                                                const float* __restrict__ init,
                                                float* __restrict__ out) {
  __shared__ float2 sA[2][CPB * TT];   // (a1,a2) per (chunk, step)
  __shared__ float  sX[2][CPB * TT];   // x per (chunk, step)

  const int base_g = blockIdx.x * CPB;
  int nActive = TOT_CH - base_g;
  if (nActive > CPB) nActive = CPB;
  const int tid = threadIdx.x;
  const bool act = tid < nActive;
  const int nElem = nActive * TT;

  int trow = 0;
  if (act) {
    int g = base_g + tid;
    int b = g / N_CH;
    int c = g - b * N_CH;
    trow = b * T_LEN + c * L_CH;
  }

  // Recurrence state registers.
  float s1 = 0.f, s2 = 0.f;              // y[t-1], y[t-2]
  float p1a = 1.f, p1b = 0.f;            // response to initial y[-1]=1
  float p2a = 0.f, p2b = 1.f;            // response to initial y[-2]=1
  if (REPLAY && act) {
    int g = base_g + tid;
    s1 = init[2 * g];
    s2 = init[2 * g + 1];
  }

  // Issue (async) staging of one time tile into LDS buffer `buf`.
  auto issue_tile = [&](int tl, int buf) {
    for (int e = tid; e < nElem; e += CPB) {
      int ci = e / TT;
      int s  = e - ci * TT;
      int g  = base_g + ci;
      int bb = g / N_CH;
      int cc = g - bb * N_CH;
      int gi = bb * T_LEN + cc * L_CH + tl * TT + s;
#if HAVE_ASYNC_LDS
      __builtin_amdgcn_global_load_async_to_lds_b64(
          (g_v2ip)(Aloop + 2 * (size_t)gi), (l_v2ip)(&sA[buf][e]), 0, 0);
      __builtin_amdgcn_global_load_async_to_lds_b32(
          (g_i32p)(xbuf + gi), (l_i32p)(&sX[buf][e]), 0, 0);
#else
      sA[buf][e] = ((const float2*)Aloop)[gi];
      sX[buf][e] = xbuf[gi];
#endif
    }
  };

  issue_tile(0, 0);

  for (int tl = 0; tl < NTILE; ++tl) {
    const int buf = tl & 1;
    wait_async_all();        // this wave's outstanding tile landed in LDS
    __syncthreads();         // ...and every other wave's too

    if (tl + 1 < NTILE) issue_tile(tl + 1, buf ^ 1);  // prefetch next tile

    if (act) {
#pragma unroll
      for (int s = 0; s < TT; ++s) {
        float2 a = sA[buf][tid * TT + s];
        float xv = sX[buf][tid * TT + s];
        if (REPLAY) {
          float yv = fmaf(-a.x, s1, fmaf(-a.y, s2, xv));
          out[trow + tl * TT + s] = yv;
          s2 = s1; s1 = yv;
        } else {
          float y0 = fmaf(-a.x, s1,  fmaf(-a.y, s2,  xv));
          float q1 = fmaf(-a.x, p1a, -a.y * p1b);
          float q2 = fmaf(-a.x, p2a, -a.y * p2b);
          s2 = s1;  s1 = y0;
          p1b = p1a; p1a = q1;
          p2b = p2a; p2a = q2;
        }
      }
    }
    __syncthreads();         // all reads of buf done before it is refilled
  }

  if (!REPLAY && act) {
    int g = base_g + tid;
    float* o = summ + 6 * (size_t)g;
    o[0] = s1;  o[1] = s2;
    o[2] = p1a; o[3] = p1b;
    o[4] = p2a; o[5] = p2b;
  }
}

// ---------------------------------------------------------------------------
// K3: serial combine across chunks of each row (tiny): propagate true states.
// ---------------------------------------------------------------------------
__global__ void k_scan_chunks(const float* __restrict__ summ,
                              float* __restrict__ init) {
  int b = blockIdx.x * blockDim.x + threadIdx.x;
  if (b >= NB) return;
  float s1 = 0.f, s2 = 0.f;
  for (int c = 0; c < N_CH; ++c) {
    int g = b * N_CH + c;
    init[2 * g]     = s1;
    init[2 * g + 1] = s2;
    const float* S = summ + 6 * (size_t)g;
    float n1 = S[0] + S[2] * s1 + S[4] * s2;
    float n2 = S[1] + S[3] * s1 + S[5] * s2;
    s1 = n1; s2 = n2;
  }
}

// ---------------------------------------------------------------------------
extern "C" void kernel_launch(void* const* d_in, const int* in_sizes, int n_in,
                              void* d_out, int out_size, void* d_ws, size_t ws_size,
                              hipStream_t stream) {
  const float* y     = (const float*)d_in[0];
  const float* Aexc  = (const float*)d_in[1];
  const float* Aloop = (const float*)d_in[2];
  float* out = (float*)d_out;

  // Workspace layout (floats): x[NB*T] | summ[TOT_CH*6] | init[TOT_CH*2]
  float* x    = (float*)d_ws;
  float* summ = x + (size_t)NB * T_LEN;
  float* init = summ + (size_t)TOT_CH * 6;

  const int nPairs = (NB * T_LEN) / 2;
  k_fir<<<(nPairs + 255) / 256, 256, 0, stream>>>(y, Aexc, x);

  const int nblk = (TOT_CH + CPB - 1) / CPB;
  k_chunks<false><<<nblk, CPB, 0, stream>>>(x, Aloop, summ, nullptr, nullptr);
  k_scan_chunks<<<1, 64, 0, stream>>>(summ, init);
  k_chunks<true><<<nblk, CPB, 0, stream>>>(x, Aloop, nullptr, init, out);
}